// LSTM_64381559767519
// MI455X (gfx1250) — compile-verified
//
#include <hip/hip_runtime.h>

// CDNA5 / gfx1250 LSTM scan:
//  - grid-persistent kernel, 512 sequential steps with device-scope barrier
//  - per-step gate GEMM done with V_WMMA_F32_16X16X4_F32 (fp32 matrix path)
//  - embedding gather fused into the K<512 half of the GEMM K-loop
//  - h_t stored directly into out[t]; c state in workspace

typedef float v2f __attribute__((ext_vector_type(2)));
typedef float v8f __attribute__((ext_vector_type(8)));

#define L_SEQ 512
#define BATCH 64
#define D_INF 512
#define HID   1024
#define KTOT  1536          // D_IN + H
#define NBLK  256           // 4 batch-row groups x 64 H-col groups
#define TPB   128           // 4 waves (wave32): one wave per gate

__device__ __forceinline__ float sigmoidf_(float x) {
  return 1.0f / (1.0f + expf(-x));
}

__global__ __launch_bounds__(TPB) void lstm_scan_kernel(
    const int*   __restrict__ x,     // [L, B]
    const float* __restrict__ emb,   // [VOCAB, D_IN]
    const float* __restrict__ pWf, const float* __restrict__ pbf,
    const float* __restrict__ pWi, const float* __restrict__ pbi,
    const float* __restrict__ pWc, const float* __restrict__ pbc,
    const float* __restrict__ pWo, const float* __restrict__ pbo,
    float* __restrict__ out,         // [L,B,H] outputs, then c[B,H], then h[B,H]
    float* __restrict__ cbuf,        // [B,H] running cell state (workspace)
    unsigned* __restrict__ cnt)      // grid barrier counter (workspace, zeroed)
{
  __shared__ float gsm[4][16][16];   // [gate][m][n] exchanged gate pre-activations

  const int lane = threadIdx.x & 31;
  const int wave = threadIdx.x >> 5;          // gate id: 0=f 1=i 2=c 3=o
  const int rb   = blockIdx.x & 3;            // batch-row group
  const int cb   = blockIdx.x >> 2;           // H-column group
  const int b0   = rb << 4;
  const int col0 = cb << 4;

  // fp32 WMMA 16x16x4 lane mapping (ISA 7.12.2):
  //   A (16x4): lanes 0-15 -> M=lane, K={0,1}; lanes 16-31 -> M=lane-16, K={2,3}
  //   B (4x16): lanes 0-15 -> N=lane, K={0,1}; lanes 16-31 -> N=lane-16, K={2,3}
  const int bl   = b0 + (lane & 15);          // batch row this lane feeds (A)
  const int koff = (lane >> 4) << 1;          // 0 or 2

  const float* Wg = (wave == 0) ? pWf : (wave == 1) ? pWi : (wave == 2) ? pWc : pWo;
  const float* wrow = Wg + (size_t)(col0 + (lane & 15)) * KTOT + koff;  // B frag row

  float* outc = out + (size_t)L_SEQ * BATCH * HID;
  float* outh = outc + (size_t)BATCH * HID;

  for (int t = 0; t < L_SEQ; ++t) {
    v8f acc = {};

    // --- K in [0, 512): z = embedding row gather ---
    {
      const int xrow = x[t * BATCH + bl];
      const float* zx = emb + (size_t)xrow * D_INF + koff;
      #pragma unroll 8
      for (int k = 0; k < D_INF; k += 4) {
        v2f a  = *(const v2f*)(zx + k);
        v2f bm = *(const v2f*)(wrow + k);
        acc = __builtin_amdgcn_wmma_f32_16x16x4_f32(
            false, a, false, bm, (short)0, acc, false, false);
      }
    }

    // --- K in [512, 1536): z = h_{t-1} (zero at t==0 -> skip whole half) ---
    if (t > 0) {
      const float* zh  = out + ((size_t)(t - 1) * BATCH + bl) * HID + koff;
      const float* wr2 = wrow + D_INF;
      #pragma unroll 8
      for (int k = 0; k < HID; k += 4) {
        v2f a  = *(const v2f*)(zh + k);
        v2f bm = *(const v2f*)(wr2 + k);
        acc = __builtin_amdgcn_wmma_f32_16x16x4_f32(
            false, a, false, bm, (short)0, acc, false, false);
      }
    }

    // --- exchange gate tiles through LDS: C/D layout VGPR r -> M = r + 8*(lane/16) ---
    {
      const int mh = (lane >> 4) << 3;
      const int n  = lane & 15;
      #pragma unroll
      for (int r = 0; r < 8; ++r) gsm[wave][mh + r][n] = acc[r];
    }
    __syncthreads();

    // --- cell update: 256 (b,h) elements, 2 per thread ---
    #pragma unroll
    for (int e = threadIdx.x; e < 256; e += TPB) {
      const int m   = e >> 4;
      const int n   = e & 15;
      const int b   = b0 + m;
      const int col = col0 + n;

      const float gf = gsm[0][m][n] + pbf[col];
      const float gi = gsm[1][m][n] + pbi[col];
      const float gc = gsm[2][m][n] + pbc[col];
      const float go = gsm[3][m][n] + pbo[col];

      const float ft = sigmoidf_(gf);
      const float it = sigmoidf_(gi);
      const float ot = sigmoidf_(go);

      const float cold = (t == 0) ? 0.0f : cbuf[(size_t)b * HID + col];
      const float cnew = ft * cold + it * tanhf(gc);
      const float h    = ot * tanhf(cnew);

      cbuf[(size_t)b * HID + col] = cnew;
      out[((size_t)t * BATCH + b) * HID + col] = h;

      if (t == L_SEQ - 1) {
        outc[(size_t)b * HID + col] = cnew;
        outh[(size_t)b * HID + col] = h;
      }
    }

    // --- device-scope barrier: all h_t visible before step t+1 reads them ---
    __syncthreads();
    if (threadIdx.x == 0) {
      __threadfence();
      __hip_atomic_fetch_add(cnt, 1u, __ATOMIC_ACQ_REL, __HIP_MEMORY_SCOPE_AGENT);
      const unsigned target = (unsigned)NBLK * (unsigned)(t + 1);
      while (__hip_atomic_load(cnt, __ATOMIC_ACQUIRE, __HIP_MEMORY_SCOPE_AGENT) < target) {
        __builtin_amdgcn_s_sleep(1);
      }
    }
    __syncthreads();
  }
}

extern "C" void kernel_launch(void* const* d_in, const int* in_sizes, int n_in,
                              void* d_out, int out_size, void* d_ws, size_t ws_size,
                              hipStream_t stream) {
  const int*   x   = (const int*)d_in[0];
  const float* emb = (const float*)d_in[1];
  const float* Wf  = (const float*)d_in[2];
  const float* bf_ = (const float*)d_in[3];
  const float* Wi  = (const float*)d_in[4];
  const float* bi_ = (const float*)d_in[5];
  const float* Wc  = (const float*)d_in[6];
  const float* bc_ = (const float*)d_in[7];
  const float* Wo  = (const float*)d_in[8];
  const float* bo_ = (const float*)d_in[9];

  float*    out  = (float*)d_out;
  unsigned* cnt  = (unsigned*)d_ws;                      // barrier counter
  float*    cbuf = (float*)((char*)d_ws + 256);          // c state [B,H]

  // zero the barrier counter region each call (graph-capture safe)
  hipMemsetAsync(d_ws, 0, 256, stream);

  lstm_scan_kernel<<<NBLK, TPB, 0, stream>>>(
      x, emb, Wf, bf_, Wi, bi_, Wc, bc_, Wo, bo_, out, cbuf, cnt);
}